// GINConv_23433341567795
// MI455X (gfx1250) — compile-verified
//
#include <hip/hip_runtime.h>

typedef __attribute__((ext_vector_type(2))) float v2f;
typedef __attribute__((ext_vector_type(8))) float v8f;

#define D_FEAT   128
#define TILE_PAD 132   // padded LDS row stride (floats): +4 banks/row -> conflict-free

__global__ __launch_bounds__(256)
void gnn_fused_wmma_kernel(const float* __restrict__ X,
                           const float* __restrict__ W,
                           const int*   __restrict__ rowptr,
                           const int*   __restrict__ colidx,
                           float*       __restrict__ out,
                           int n_nodes)
{
    // Aggregated 16 x 128 fp32 tile (padded) + transposed weights sWt[n][k]
    __shared__ float sA [16 * TILE_PAD];
    __shared__ float sWt[D_FEAT * TILE_PAD];

    const int tid  = threadIdx.x;
    const int lane = tid & 31;
    const int wv   = tid >> 5;            // wave id 0..7
    const int r0   = blockIdx.x * 16;     // first node row of this window

    // ---- Phase 0: stage W transposed into LDS (coalesced read, ~2-way write) ----
    for (int i = tid; i < D_FEAT * D_FEAT; i += 256) {
        int k = i >> 7;        // W row (input feature)
        int n = i & 127;       // W col (output feature)
        sWt[n * TILE_PAD + k] = W[i];
    }

    // ---- Phase 1: gather + segment-sum; wave wv aggregates rows 2wv, 2wv+1 ----
    for (int rr = 0; rr < 2; ++rr) {
        const int rl  = wv * 2 + rr;      // local row 0..15
        const int row = r0 + rl;
        float4 acc = make_float4(0.f, 0.f, 0.f, 0.f);
        if (row < n_nodes) {
            const int e0 = rowptr[row];
            const int e1 = rowptr[row + 1];
            for (int eb = e0; eb < e1; eb += 32) {
                int ci = 0;
                if (eb + lane < e1) ci = colidx[eb + lane];   // coalesced index load
                const int cnt = min(32, e1 - eb);
                for (int j = 0; j < cnt; ++j) {
                    const int c = __shfl(ci, j, 32);          // broadcast edge column
                    // full 512B source row loaded coalesced across the wave
                    const float4 v =
                        *(const float4*)(X + (size_t)c * D_FEAT + lane * 4);
                    acc.x += v.x; acc.y += v.y; acc.z += v.z; acc.w += v.w;
                }
            }
        }
        *(float4*)(sA + rl * TILE_PAD + lane * 4) = acc;      // zeros for OOB rows
    }

    __syncthreads();

    // ---- Phase 2: fp32 WMMA GEMM: (16x128 tile) x (128x16 W-slice) per wave ----
    // A frag (16x4 f32): lanes 0-15 -> M=lane, K={kb,kb+1}; lanes 16-31 -> K={kb+2,kb+3}
    // B frag (4x16 f32): mirrored on N; both are contiguous b64 LDS loads.
    const int n0   = wv * 16;             // output-column base for this wave
    const int mrow = lane & 15;
    const int kd   = (lane >> 4) * 2;     // 0 or 2

    v8f c = {};
    const float* aBase = sA  + mrow * TILE_PAD + kd;
    const float* bBase = sWt + (n0 + mrow) * TILE_PAD + kd;
    #pragma unroll
    for (int kk = 0; kk < D_FEAT / 4; ++kk) {
        v2f a = *(const v2f*)(aBase + kk * 4);
        v2f b = *(const v2f*)(bBase + kk * 4);
        // D = A*B + C, full fp32 (matches reference precision)
        c = __builtin_amdgcn_wmma_f32_16x16x4_f32(
                /*neg_a=*/false, a, /*neg_b=*/false, b,
                /*c_mod=*/(short)0, c, /*reuse_a=*/false, /*reuse_b=*/false);
    }

    // ---- Phase 3: store D per 16x16 f32 C/D layout ----
    const int ncol  = n0 + (lane & 15);
    const int mbase = (lane >> 4) * 8;    // lanes 16-31 hold M=v+8
    #pragma unroll
    for (int v = 0; v < 8; ++v) {
        const int row = r0 + mbase + v;
        if (row < n_nodes)
            out[(size_t)row * D_FEAT + ncol] = c[v];
    }
}

extern "C" void kernel_launch(void* const* d_in, const int* in_sizes, int n_in,
                              void* d_out, int out_size, void* d_ws, size_t ws_size,
                              hipStream_t stream) {
    const float* X      = (const float*)d_in[0];   // [N, 128]
    const float* W      = (const float*)d_in[1];   // [128, 128]
    const int*   rowptr = (const int*)  d_in[2];   // [N+1]
    const int*   colidx = (const int*)  d_in[3];   // [E]
    float*       out    = (float*)d_out;           // [N, 128]

    const int n_nodes   = in_sizes[2] - 1;
    const int n_windows = (n_nodes + 15) / 16;

    hipLaunchKernelGGL(gnn_fused_wmma_kernel,
                       dim3(n_windows), dim3(256), 0, stream,
                       X, W, rowptr, colidx, out, n_nodes);
}